// CRF_59390807769482
// MI455X (gfx1250) — compile-verified
//
#include <hip/hip_runtime.h>
#include <hip/hip_bf16.h>

#define B_    128
#define T_    512
#define TAG_  256
#define START_ (TAG_ - 2)
#define STOP_  (TAG_ - 1)

// ---- CDNA5 async global->LDS data mover (inline asm; portable across both
// toolchains since the clang builtin arity differs). GVS addressing:
// mem_addr = SADDR(64b sgpr) + VADDR(32b vgpr); dest LDS byte addr in VDST vgpr.
__device__ __forceinline__ void async_ld_f32(unsigned lds_addr, unsigned voff,
                                             const float* sbase) {
  asm volatile("global_load_async_to_lds_b32 %0, %1, %2"
               :: "v"(lds_addr), "v"(voff), "s"(sbase)
               : "memory");
}
__device__ __forceinline__ void wait_async0() {
  asm volatile("s_wait_asynccnt 0" ::: "memory");
}

// Forward Viterbi: one block per batch element, thread j owns tag-column j.
// LDS: transitions (256KB, CDNA5 320KB/WGP makes this fit) + partition (1KB)
//      + emit double buffer (2KB).
extern "C" __global__ void __launch_bounds__(TAG_)
viterbi_fwd(const float* __restrict__ feats, const int* __restrict__ mask,
            const float* __restrict__ trans, float* __restrict__ score_out,
            int* __restrict__ ptr_out, unsigned char* __restrict__ bp) {
  extern __shared__ float smem[];
  float* sTrans = smem;                 // TAG_*TAG_ floats
  float* sPart  = smem + TAG_ * TAG_;   // TAG_ floats
  float* sEmit  = sPart + TAG_;         // 2*TAG_ floats (double buffer)

  const int b = blockIdx.x;
  const int j = threadIdx.x;

  // Stage full transition matrix into LDS, 16B-vectorized:
  // 16384 float4s / 256 threads = 64 b128 loads + 64 ds_store_b128 per thread.
  {
    const float4* t4 = (const float4*)trans;
    float4*       s4 = (float4*)sTrans;
#pragma unroll 8
    for (int k = 0; k < (TAG_ * TAG_ / 4) / TAG_; ++k)   // 64 iters
      s4[k * TAG_ + j] = t4[k * TAG_ + j];
  }
  __syncthreads();

  const float* frow = feats + (size_t)b * T_ * TAG_;

  // part0 = feats[b,0,:] + trans[START,:]
  sPart[j] = frow[j] + sTrans[START_ * TAG_ + j];

  // Prefetch emit row t=1 into buffer (1&1)=1 via the async data mover.
  {
    unsigned lds = (unsigned)(size_t)&sEmit[1 * TAG_ + j];
    async_ld_f32(lds, j * 4u, frow + (size_t)1 * TAG_);
  }

  for (int t = 1; t < T_; ++t) {
    const int cur = t & 1;
    wait_async0();        // per-wave: our async emit row has landed in LDS
    __syncthreads();      // all waves' emit data + previous sPart update visible

    // Kick off prefetch of emit[t+1] into the other buffer while computing.
    if (t + 1 < T_) {
      unsigned lds = (unsigned)(size_t)&sEmit[(1 - cur) * TAG_ + j];
      async_ld_f32(lds, j * 4u, frow + (size_t)(t + 1) * TAG_);
    }

    // Tropical mat-vec: best over i of part[i] + trans[i][j].
    // Strict '>' keeps the first max -> matches jnp.argmax tie-breaking.
    float best = -__builtin_inff();
    int   bi   = 0;
    const float*  col = sTrans + j;
    const float4* p4  = (const float4*)sPart;
#pragma unroll 4
    for (int i4 = 0; i4 < TAG_ / 4; ++i4) {
      const float4 p = p4[i4];                 // one ds_load_b128, broadcast
      const int    i = i4 * 4;
      float v0 = p.x + col[(i + 0) * TAG_];
      float v1 = p.y + col[(i + 1) * TAG_];
      float v2 = p.z + col[(i + 2) * TAG_];
      float v3 = p.w + col[(i + 3) * TAG_];
      if (v0 > best) { best = v0; bi = i; }
      if (v1 > best) { best = v1; bi = i + 1; }
      if (v2 > best) { best = v2; bi = i + 2; }
      if (v3 > best) { best = v3; bi = i + 3; }
    }

    const int m = mask[b * T_ + t];            // uniform -> scalar load
    bp[((size_t)b * (T_ - 1) + (t - 1)) * TAG_ + j] =
        (unsigned char)(m > 0 ? bi : 0);

    __syncthreads();      // everyone done reading sPart
    sPart[j] = best + sEmit[cur * TAG_ + j];
  }
  __syncthreads();

  // Termination: path_vals[j] = max_i(part[i] + trans[i][j]); need column STOP.
  float best = -__builtin_inff();
  int   bi   = 0;
  {
    const float*  col = sTrans + j;
    const float4* p4  = (const float4*)sPart;
#pragma unroll 4
    for (int i4 = 0; i4 < TAG_ / 4; ++i4) {
      const float4 p = p4[i4];
      const int    i = i4 * 4;
      float v0 = p.x + col[(i + 0) * TAG_];
      float v1 = p.y + col[(i + 1) * TAG_];
      float v2 = p.z + col[(i + 2) * TAG_];
      float v3 = p.w + col[(i + 3) * TAG_];
      if (v0 > best) { best = v0; bi = i; }
      if (v1 > best) { best = v1; bi = i + 1; }
      if (v2 > best) { best = v2; bi = i + 2; }
      if (v3 > best) { best = v3; bi = i + 3; }
    }
  }
  if (j == STOP_) {
    score_out[b] = best;   // path_score[b]
    ptr_out[b]   = bi;     // initial backtrace pointer
  }
}

// Backtrace: one lane per batch; bp array (16MB) is fully L2-resident (192MB L2).
extern "C" __global__ void viterbi_bt(const unsigned char* __restrict__ bp,
                                      const int* __restrict__ ptr_in,
                                      float* __restrict__ dec_out) {
  int b = threadIdx.x + blockIdx.x * blockDim.x;
  if (b >= B_) return;
  int ptr = ptr_in[b];
  float* dec = dec_out + (size_t)b * T_;
  dec[T_ - 1] = (float)ptr;
  for (int t = T_ - 2; t >= 0; --t) {
    // decode[t] = bp_time(t+1)[decode[t+1]]; bp_time(s) stored at slot s-1.
    ptr = bp[((size_t)b * (T_ - 1) + t) * TAG_ + ptr];
    dec[t] = (float)ptr;
  }
}

extern "C" void kernel_launch(void* const* d_in, const int* in_sizes, int n_in,
                              void* d_out, int out_size, void* d_ws, size_t ws_size,
                              hipStream_t stream) {
  const float* feats = (const float*)d_in[0];   // (128,512,256) f32
  const int*   mask  = (const int*)d_in[1];     // (128,512) i32
  const float* trans = (const float*)d_in[2];   // (256,256) f32
  float* out = (float*)d_out;                   // [0..127]=score, then decode (B,T) as float

  // Workspace: backpointers (B*(T-1)*TAG u8 = ~16MB) then pointer[B] i32.
  unsigned char* bpw = (unsigned char*)d_ws;
  const size_t bpBytes = (size_t)B_ * (T_ - 1) * TAG_;
  int* ptrws = (int*)((char*)d_ws + bpBytes);

  const size_t shmem = ((size_t)TAG_ * TAG_ + TAG_ + 2 * TAG_) * sizeof(float);
  viterbi_fwd<<<B_, TAG_, shmem, stream>>>(feats, mask, trans, out, ptrws, bpw);
  viterbi_bt<<<1, B_, 0, stream>>>(bpw, ptrws, out + B_);
}